// linear_dot_product_attnention_90074054132189
// MI455X (gfx1250) — compile-verified
//
#include <hip/hip_runtime.h>

typedef __attribute__((ext_vector_type(2))) float v2f;
typedef __attribute__((ext_vector_type(8))) float v8f;

#define B_   4
#define S_   4096
#define H_   16
#define D_   64
#define C_   128
#define N_   32            // S_/C_
#define BH_  (B_*H_)
#define RS_  (H_*D_)       // 1024 floats between consecutive s rows in [B,S,H,D]
#define PAD_ 68            // padded LDS row stride (mult of 4 -> float4 stores, bank stride 4)
#define KVSZ_ (D_*D_)      // 4096

__device__ __forceinline__ v8f wmma_f32(v2f a, v2f b, v8f c) {
  // D = A(16x4, f32) * B(4x16, f32) + C(16x16, f32)
  return __builtin_amdgcn_wmma_f32_16x16x4_f32(false, a, false, b, (short)0, c, false, false);
}

// ---------------------------------------------------------------------------
// Kernel 1: per-chunk KV outer product  kv[d][e] = sum_j K[j][d] * V[j][e]
// grid = B*H*N blocks, 512 threads (16 waves); one 16x16 tile per wave.
// ---------------------------------------------------------------------------
__global__ __launch_bounds__(512) void kv_chunk_kernel(const float* __restrict__ k,
                                                       const float* __restrict__ v,
                                                       float* __restrict__ kvws) {
  __shared__ float sK[C_ * PAD_];
  __shared__ float sV[C_ * PAD_];
  const int tid = threadIdx.x;
  const int blk = blockIdx.x;
  const int bh = blk / N_, n = blk % N_;
  const int b = bh / H_, h = bh % H_;
  const size_t base0 = ((size_t)(b * S_ + n * C_) * H_ + h) * D_;

  const int lr = tid >> 4, lc = (tid & 15) << 2;
  for (int it = 0; it < 4; ++it) {
    int row = lr + it * 32;
    size_t g = base0 + (size_t)row * RS_ + lc;
    *(float4*)&sK[row * PAD_ + lc] = *(const float4*)(k + g);
    *(float4*)&sV[row * PAD_ + lc] = *(const float4*)(v + g);
  }
  __syncthreads();

  const int w = tid >> 5;            // 0..15
  const int lane = tid & 31;
  const int half = lane >> 4;
  const int lrow = lane & 15;
  const int db = w >> 2, eb = w & 3; // 4x4 tile grid, 1 tile per wave
  float* out = kvws + (size_t)blk * KVSZ_;

  v8f acc = {};
  for (int jk = 0; jk < C_; jk += 4) {
    const int jr = jk + 2 * half;
    v2f a, bb;
    a.x = sK[jr * PAD_ + db * 16 + lrow];        // A = K^T : A[d][j] = K[j][d]
    a.y = sK[(jr + 1) * PAD_ + db * 16 + lrow];
    bb.x = sV[jr * PAD_ + eb * 16 + lrow];       // B = V
    bb.y = sV[(jr + 1) * PAD_ + eb * 16 + lrow];
    acc = wmma_f32(a, bb, acc);
  }
#pragma unroll
  for (int r = 0; r < 8; ++r)
    out[(db * 16 + r + 8 * half) * D_ + eb * 16 + lrow] = acc[r];
}

// ---------------------------------------------------------------------------
// Kernel 2: in-place exclusive prefix sum over chunks of the KV states.
// grid = BH*16 blocks, 256 threads; one thread per (bh, d*64+e) element.
// ---------------------------------------------------------------------------
__global__ __launch_bounds__(256) void kv_prefix_kernel(float* __restrict__ kvws) {
  const int de = (blockIdx.x & 15) * 256 + threadIdx.x;  // 0..4095
  const int bh = blockIdx.x >> 4;
  float* base = kvws + (size_t)bh * N_ * KVSZ_ + de;
  float prev = 0.f;
  for (int nn = 0; nn < N_; ++nn) {
    float t = base[(size_t)nn * KVSZ_];
    base[(size_t)nn * KVSZ_] = prev;
    prev += t;
  }
}

// ---------------------------------------------------------------------------
// Kernel 3: per chunk  out = tril(Q K^T) V + Q KV_prev.
// grid = B*H*N blocks, 512 threads (16 waves).
// Phase A: 36 causal score tiles round-robined over waves (balanced).
// Phase B: 32 output tiles round-robined over waves (balanced).
// ---------------------------------------------------------------------------
__global__ __launch_bounds__(512) void chunk_attn_kernel(const float* __restrict__ q,
                                                         const float* __restrict__ k,
                                                         const float* __restrict__ v,
                                                         const float* __restrict__ kvws,
                                                         float* __restrict__ out) {
  __shared__ float sQ[C_ * PAD_];
  __shared__ float sK[C_ * PAD_];
  __shared__ float sV[C_ * PAD_];
  __shared__ float sKV[KVSZ_];
  __shared__ float sS[36 * 288];  // score tile t=(mb*(mb+1)/2+jb): 16 rows x stride 18

  const int tid = threadIdx.x;
  const int blk = blockIdx.x;
  const int bh = blk / N_, n = blk % N_;
  const int b = bh / H_, h = bh % H_;
  const size_t base0 = ((size_t)(b * S_ + n * C_) * H_ + h) * D_;

  const int lr = tid >> 4, lc = (tid & 15) << 2;
  for (int it = 0; it < 4; ++it) {
    int row = lr + it * 32;
    size_t g = base0 + (size_t)row * RS_ + lc;
    *(float4*)&sQ[row * PAD_ + lc] = *(const float4*)(q + g);
    *(float4*)&sK[row * PAD_ + lc] = *(const float4*)(k + g);
    *(float4*)&sV[row * PAD_ + lc] = *(const float4*)(v + g);
  }
  const float* kvp = kvws + (size_t)blk * KVSZ_;
  for (int i = tid * 4; i < KVSZ_; i += 2048)
    *(float4*)&sKV[i] = *(const float4*)(kvp + i);
  __syncthreads();

  const int w = tid >> 5;   // 0..15
  const int lane = tid & 31;
  const int half = lane >> 4;
  const int lrow = lane & 15;

  // ---- Phase A: score tiles S(mb,jb) = Q_mb K_jb^T, jb <= mb, causal-masked
  for (int t = w; t < 36; t += 16) {
    int mb = 0;
    while (((mb + 1) * (mb + 2)) >> 1 <= t) ++mb;   // uniform per wave
    const int jb = t - ((mb * (mb + 1)) >> 1);
    v8f acc = {};
    const int arow = (mb * 16 + lrow) * PAD_;
    const int brow = (jb * 16 + lrow) * PAD_;
#pragma unroll
    for (int kk = 0; kk < D_; kk += 4) {
      v2f a = *(const v2f*)&sQ[arow + kk + 2 * half];   // A = Q
      v2f bb = *(const v2f*)&sK[brow + kk + 2 * half];  // B = K^T : B[kidx][j] = K[j][kidx]
      acc = wmma_f32(a, bb, acc);
    }
    float* st = &sS[t * 288];
    if (jb == mb) {
#pragma unroll
      for (int r = 0; r < 8; ++r) {
        const int mi = r + 8 * half;
        st[mi * 18 + lrow] = (lrow <= mi) ? acc[r] : 0.f;  // causal mask on diagonal tile
      }
    } else {
#pragma unroll
      for (int r = 0; r < 8; ++r)
        st[(r + 8 * half) * 18 + lrow] = acc[r];
    }
  }
  __syncthreads();

  // ---- Phase B: output tiles (mb, eb) = sum_{jb<=mb} S V + Q KV_prev
  for (int t2 = w; t2 < 32; t2 += 16) {
    const int mb = t2 >> 2, eb = t2 & 3;
    const int arow = (mb * 16 + lrow) * PAD_;
    v8f acc = {};
#pragma unroll
    for (int kk = 0; kk < D_; kk += 4) {               // inter-chunk: Q @ KV_prev
      v2f a = *(const v2f*)&sQ[arow + kk + 2 * half];
      const int kr = kk + 2 * half;
      v2f bb;
      bb.x = sKV[kr * D_ + eb * 16 + lrow];
      bb.y = sKV[(kr + 1) * D_ + eb * 16 + lrow];
      acc = wmma_f32(a, bb, acc);
    }
    const int sbase = ((mb * (mb + 1)) >> 1) * 288;
    for (int jb = 0; jb <= mb; ++jb) {                 // intra-chunk: S @ V
      const float* st = &sS[sbase + jb * 288];
#pragma unroll
      for (int jj = 0; jj < 16; jj += 4) {
        v2f a = *(const v2f*)&st[lrow * 18 + jj + 2 * half];
        const int jr = jb * 16 + jj + 2 * half;
        v2f bb;
        bb.x = sV[jr * PAD_ + eb * 16 + lrow];
        bb.y = sV[(jr + 1) * PAD_ + eb * 16 + lrow];
        acc = wmma_f32(a, bb, acc);
      }
    }
#pragma unroll
    for (int r = 0; r < 8; ++r) {
      const int row = mb * 16 + r + 8 * half;
      out[base0 + (size_t)row * RS_ + eb * 16 + lrow] = acc[r];
    }
  }
}

// ---------------------------------------------------------------------------
extern "C" void kernel_launch(void* const* d_in, const int* in_sizes, int n_in,
                              void* d_out, int out_size, void* d_ws, size_t ws_size,
                              hipStream_t stream) {
  (void)in_sizes; (void)n_in; (void)out_size; (void)ws_size;
  const float* q = (const float*)d_in[0];
  const float* k = (const float*)d_in[1];
  const float* v = (const float*)d_in[2];
  // d_in[3] = attn_mask: unused by the reference forward.
  float* out = (float*)d_out;
  float* kvws = (float*)d_ws;  // needs B*H*N*D*D*4 = 32 MB

  kv_chunk_kernel<<<BH_ * N_, 512, 0, stream>>>(k, v, kvws);
  kv_prefix_kernel<<<BH_ * 16, 256, 0, stream>>>(kvws);
  chunk_attn_kernel<<<BH_ * N_, 512, 0, stream>>>(q, k, v, kvws, out);
}